// SimpleRGCN_7318624272855
// MI455X (gfx1250) — compile-verified
//
#include <hip/hip_runtime.h>
#include <hip/hip_bf16.h>

typedef __attribute__((ext_vector_type(2))) float v2f;
typedef __attribute__((ext_vector_type(8))) float v8f;
typedef __attribute__((ext_vector_type(4))) int   v4i;

#define LN_EPS 1e-5f

#if defined(__has_builtin)
#if __has_builtin(__builtin_amdgcn_global_load_async_to_lds_b128) && \
    __has_builtin(__builtin_amdgcn_s_wait_asynccnt)
#define HAVE_ASYNC_LDS 1
#endif
#endif
#ifndef HAVE_ASYNC_LDS
#define HAVE_ASYNC_LDS 0
#endif

// D = 64 (node feature dim), H = 128 (hidden dim) are hard-wired in tile code.

static __device__ __forceinline__ v8f wmma_f32_16x16x4(v2f a, v2f b, v8f c) {
    return __builtin_amdgcn_wmma_f32_16x16x4_f32(
        /*neg_a=*/false, a, /*neg_b=*/false, b,
        /*c_mod=*/(short)0, c, /*reuse_a=*/false, /*reuse_b=*/false);
}

// 16B global -> LDS copy; async (ASYNCcnt) path when the toolchain exposes it.
static __device__ __forceinline__ void copy_b128_to_lds(const float* gsrc, float* ldst) {
#if HAVE_ASYNC_LDS
    __builtin_amdgcn_global_load_async_to_lds_b128((v4i*)gsrc, (v4i*)ldst, 0, 0);
#else
    *(float4*)ldst = *(const float4*)gsrc;
#endif
}
static __device__ __forceinline__ void copy_join() {
#if HAVE_ASYNC_LDS
    __builtin_amdgcn_s_wait_asynccnt(0);
#endif
}

__global__ void fill_zero_kernel(float* __restrict__ p, long n) {
    long i = (long)blockIdx.x * blockDim.x + threadIdx.x;
    long stride = (long)gridDim.x * blockDim.x;
    for (; i < n; i += stride) p[i] = 0.0f;
}

// acc1[r, dst, :] += emb[node_ids[src], :]   (one thread per (edge, float4 chunk))
__global__ void rgcn_scatter1(const int* __restrict__ node_ids,
                              const int* __restrict__ edge_index,
                              const int* __restrict__ edge_type,
                              const float* __restrict__ emb,
                              float* __restrict__ acc, int N, int E) {
    long t = (long)blockIdx.x * blockDim.x + threadIdx.x;
    if (t >= (long)E * 16) return;
    int e  = (int)(t >> 4);
    int c4 = (int)(t & 15);
    int src = edge_index[e];
    int dst = edge_index[E + e];
    int r   = edge_type[e];
    float4 v = *(const float4*)(emb + (long)node_ids[src] * 64 + c4 * 4);
    float* base = acc + ((long)r * N + dst) * 64 + c4 * 4;
    atomicAdd(base + 0, v.x);
    atomicAdd(base + 1, v.y);
    atomicAdd(base + 2, v.z);
    atomicAdd(base + 3, v.w);
}

// x1 = relu(LN(sum_r acc1_r @ W1_r))    block: 16 rows x 128 cols, 8 waves
__global__ __launch_bounds__(256) void rgcn_gemm1_ln_relu(
        const float* __restrict__ acc,   // [R, N, 64]
        const float* __restrict__ W1,    // [R, 64, 128]
        const float* __restrict__ g1, const float* __restrict__ b1,
        float* __restrict__ x1,          // [N, 128]
        int N, int R) {
    __shared__ float lA[16 * 68];              // A tile, padded stride 68
    __shared__ float lB[16 * 256 * 2 + 32];    // pair-interleaved W panel (even/odd split)
    __shared__ float lC[16 * 128];             // output staging for fused LN

    float* lBE = lB;                       // pairs (4q, 4q+1)   -> lanes 0-15
    float* lBO = lB + 16 * 256 + 32;       // pairs (4q+2, 4q+3) -> lanes 16-31 (bank-shifted)

    const int tid  = threadIdx.x;
    const int wave = tid >> 5;
    const int lane = tid & 31;
    const int row0 = blockIdx.x * 16;
    const int col0 = wave * 16;

    v8f c = {0.f, 0.f, 0.f, 0.f, 0.f, 0.f, 0.f, 0.f};

    const int mrow = lane & 15;
    const int koff = (lane < 16) ? 0 : 2;
    const float* aBase = lA + mrow * 68 + koff;
    const float* bBase = ((lane < 16) ? lBE : lBO) + (col0 + mrow) * 2;

    for (int r = 0; r < R; ++r) {
        {   // stage A: 16 contiguous rows x 64 f32 -> async b128 per thread
            int arow = tid >> 4, ac4 = tid & 15;
            copy_b128_to_lds(acc + ((long)r * N + row0 + arow) * 64 + ac4 * 4,
                             lA + arow * 68 + ac4 * 4);
        }
        {   // stage B: W1[r] 64x128 -> pair-interleaved (2x4 mini-blocks, 4 per thread)
            const float* Wr = W1 + (long)r * 64 * 128;
#pragma unroll
            for (int i = 0; i < 4; ++i) {
                int mb  = tid + 256 * i;        // 0..1023
                int q   = mb >> 6;              // 0..15
                int par = (mb >> 5) & 1;
                int cq  = mb & 31;              // column quad
                int k0  = 4 * q + 2 * par;
                float4 r0 = *(const float4*)(Wr + k0 * 128 + cq * 4);
                float4 r1 = *(const float4*)(Wr + (k0 + 1) * 128 + cq * 4);
                float2* d2 = (float2*)((par ? lBO : lBE) + q * 256 + cq * 8);
                d2[0] = make_float2(r0.x, r1.x);
                d2[1] = make_float2(r0.y, r1.y);
                d2[2] = make_float2(r0.z, r1.z);
                d2[3] = make_float2(r0.w, r1.w);
            }
        }
        copy_join();
        __syncthreads();
        if (r + 1 < R)   // pull next relation's panel toward the WGP
            __builtin_prefetch(W1 + (long)(r + 1) * 64 * 128 + tid * 32, 0, 0);

#pragma unroll
        for (int q = 0; q < 16; ++q) {          // k = 4q
            v2f a = *(const v2f*)(aBase + 4 * q);
            v2f b = *(const v2f*)(bBase + 256 * q);
            c = wmma_f32_16x16x4(a, b, c);
        }
        __syncthreads();
    }

    // stage C tiles: VGPR i holds rows M=i (lanes 0-15) and M=i+8 (lanes 16-31)
    const int mbase = (lane < 16) ? 0 : 8;
#pragma unroll
    for (int i = 0; i < 8; ++i)
        lC[(mbase + i) * 128 + col0 + mrow] = c[i];
    __syncthreads();

    // fused LayerNorm + ReLU over the 16 complete rows (reuse lA as scratch)
    const int row = tid >> 4, sub = tid & 15;
    float s = 0.f, q2 = 0.f;
#pragma unroll
    for (int j = 0; j < 8; ++j) {
        float v = lC[row * 128 + sub + 16 * j];
        s += v; q2 += v * v;
    }
    lA[row * 16 + sub] = s;
    lA[256 + row * 16 + sub] = q2;
    __syncthreads();
    if (sub == 0) {
        float S = 0.f, Q = 0.f;
#pragma unroll
        for (int j = 0; j < 16; ++j) { S += lA[row * 16 + j]; Q += lA[256 + row * 16 + j]; }
        float mu  = S * (1.f / 128.f);
        float var = Q * (1.f / 128.f) - mu * mu;
        lA[512 + row] = mu;
        lA[528 + row] = rsqrtf(var + LN_EPS);
    }
    __syncthreads();
    float mu = lA[512 + row], rs = lA[528 + row];
#pragma unroll
    for (int j = 0; j < 8; ++j) {
        int col = sub + 16 * j;
        float v = (lC[row * 128 + col] - mu) * rs * g1[col] + b1[col];
        x1[(long)(row0 + row) * 128 + col] = fmaxf(v, 0.f);
    }
}

// xr2[r] = x1 @ W2[r]    block: 16 rows x 64 cols of one relation, 4 waves
__global__ __launch_bounds__(128) void rgcn_gemm2(
        const float* __restrict__ x1,    // [N, 128]
        const float* __restrict__ W2,    // [R, 128, 64]
        float* __restrict__ xr2,         // [R, N, 64]
        int N) {
    __shared__ float lA[16 * 132];             // 16 x 128, padded stride 132
    __shared__ float lB[32 * 128 * 2 + 32];    // pair-interleaved W2[r] (even/odd split)

    float* lBE = lB;
    float* lBO = lB + 32 * 128 + 32;

    const int tid  = threadIdx.x;
    const int wave = tid >> 5;
    const int lane = tid & 31;
    const int row0 = blockIdx.x * 16;
    const int r    = blockIdx.y;
    const int col0 = wave * 16;

    {   // stage A: 16 x 128 f32 (512 x b128, 4 per thread, async)
#pragma unroll
        for (int i = 0; i < 4; ++i) {
            int idx  = tid + 128 * i;
            int arow = idx >> 5, ac4 = idx & 31;
            copy_b128_to_lds(x1 + (long)(row0 + arow) * 128 + ac4 * 4,
                             lA + arow * 132 + ac4 * 4);
        }
    }
    {   // stage B: W2[r] 128x64 -> pair-interleaved (1024 mini-blocks, 8 per thread)
        const float* Wr = W2 + (long)r * 128 * 64;
#pragma unroll
        for (int i = 0; i < 8; ++i) {
            int mb  = tid + 128 * i;        // 0..1023
            int q   = mb >> 5;              // 0..31
            int par = (mb >> 4) & 1;
            int cq  = mb & 15;              // column quad
            int k0  = 4 * q + 2 * par;
            float4 r0 = *(const float4*)(Wr + k0 * 64 + cq * 4);
            float4 r1 = *(const float4*)(Wr + (k0 + 1) * 64 + cq * 4);
            float2* d2 = (float2*)((par ? lBO : lBE) + q * 128 + cq * 8);
            d2[0] = make_float2(r0.x, r1.x);
            d2[1] = make_float2(r0.y, r1.y);
            d2[2] = make_float2(r0.z, r1.z);
            d2[3] = make_float2(r0.w, r1.w);
        }
    }
    copy_join();
    __syncthreads();

    v8f c = {0.f, 0.f, 0.f, 0.f, 0.f, 0.f, 0.f, 0.f};
    const int mrow = lane & 15;
    const int koff = (lane < 16) ? 0 : 2;
    const float* aBase = lA + mrow * 132 + koff;
    const float* bBase = ((lane < 16) ? lBE : lBO) + (col0 + mrow) * 2;
#pragma unroll
    for (int q = 0; q < 32; ++q) {              // k = 4q
        v2f a = *(const v2f*)(aBase + 4 * q);
        v2f b = *(const v2f*)(bBase + 128 * q);
        c = wmma_f32_16x16x4(a, b, c);
    }

    const int mbase = (lane < 16) ? 0 : 8;
#pragma unroll
    for (int i = 0; i < 8; ++i)
        xr2[((long)r * N + row0 + mbase + i) * 64 + col0 + mrow] = c[i];
}

// out[dst, :] += xr2[edge_type, src, :]   (one thread per (edge, float4 chunk))
__global__ void rgcn_scatter2(const int* __restrict__ edge_index,
                              const int* __restrict__ edge_type,
                              const float* __restrict__ xr2,
                              float* __restrict__ out, int N, int E) {
    long t = (long)blockIdx.x * blockDim.x + threadIdx.x;
    if (t >= (long)E * 16) return;
    int e  = (int)(t >> 4);
    int c4 = (int)(t & 15);
    int src = edge_index[e];
    int dst = edge_index[E + e];
    int r   = edge_type[e];
    float4 v = *(const float4*)(xr2 + ((long)r * N + src) * 64 + c4 * 4);
    float* base = out + (long)dst * 64 + c4 * 4;
    atomicAdd(base + 0, v.x);
    atomicAdd(base + 1, v.y);
    atomicAdd(base + 2, v.z);
    atomicAdd(base + 3, v.w);
}

// in-place: out = LN(out)*g2 + b2 + emb[node_ids]
__global__ __launch_bounds__(64) void rgcn_ln2_residual(
        float* __restrict__ out, const float* __restrict__ emb,
        const int* __restrict__ node_ids,
        const float* __restrict__ g2, const float* __restrict__ b2) {
    __shared__ float ss[64];
    __shared__ float sq[64];
    const int n = blockIdx.x, t = threadIdx.x;
    float v = out[(long)n * 64 + t];
    ss[t] = v; sq[t] = v * v;
    __syncthreads();
    for (int s = 32; s > 0; s >>= 1) {
        if (t < s) { ss[t] += ss[t + s]; sq[t] += sq[t + s]; }
        __syncthreads();
    }
    float mu  = ss[0] * (1.f / 64.f);
    float var = sq[0] * (1.f / 64.f) - mu * mu;
    float rs  = rsqrtf(var + LN_EPS);
    out[(long)n * 64 + t] = (v - mu) * rs * g2[t] + b2[t]
                          + emb[(long)node_ids[n] * 64 + t];
}

extern "C" void kernel_launch(void* const* d_in, const int* in_sizes, int n_in,
                              void* d_out, int out_size, void* d_ws, size_t ws_size,
                              hipStream_t stream) {
    const int*   node_ids   = (const int*)d_in[0];
    const int*   edge_index = (const int*)d_in[1];
    const int*   edge_type  = (const int*)d_in[2];
    const float* emb        = (const float*)d_in[3];
    const float* W1         = (const float*)d_in[4];
    const float* W2         = (const float*)d_in[5];
    const float* g1         = (const float*)d_in[6];
    const float* b1         = (const float*)d_in[7];
    const float* g2         = (const float*)d_in[8];
    const float* b2         = (const float*)d_in[9];

    const int N = in_sizes[0];                  // 50000
    const int E = in_sizes[2];                  // 600000
    const int H = in_sizes[6];                  // 128
    const int D = in_sizes[3] / N;              // 64
    const int R = in_sizes[4] / (D * H);        // 16

    float* out  = (float*)d_out;
    float* acc1 = (float*)d_ws;                       // [R,N,D] — reused as xr2
    float* x1   = acc1 + (size_t)R * N * D;           // [N,H]

    const long accElems = (long)R * N * D;

    // 1) zero acc1 and d_out
    fill_zero_kernel<<<4096, 256, 0, stream>>>(acc1, accElems);
    fill_zero_kernel<<<1024, 256, 0, stream>>>(out, (long)N * D);

    // 2) layer-1 aggregate: acc1[r,dst,:] += x[src,:]
    long sthreads = (long)E * 16;
    rgcn_scatter1<<<(unsigned)((sthreads + 255) / 256), 256, 0, stream>>>(
        node_ids, edge_index, edge_type, emb, acc1, N, E);

    // 3) x1 = relu(LN(sum_r acc1_r @ W1_r))
    rgcn_gemm1_ln_relu<<<N / 16, 256, 0, stream>>>(acc1, W1, g1, b1, x1, N, R);

    // 4) layer-2 transform: xr2[r] = x1 @ W2[r]  (overwrites acc1 region)
    rgcn_gemm2<<<dim3(N / 16, R), 128, 0, stream>>>(x1, W2, acc1, N);

    // 5) layer-2 aggregate into d_out
    rgcn_scatter2<<<(unsigned)((sthreads + 255) / 256), 256, 0, stream>>>(
        edge_index, edge_type, acc1, out, N, E);

    // 6) LN + residual in place
    rgcn_ln2_residual<<<N, 64, 0, stream>>>(out, emb, node_ids, g2, b2);
}